// FeatureLayer_69604239999291
// MI455X (gfx1250) — compile-verified
//
#include <hip/hip_runtime.h>

// Embedding-bag feature layer for MI455X (gfx1250, wave32).
// One wave32 per sample row. Gather/L2-bandwidth bound.
// Strategy: wide scalar (SMEM) index loads with a one-chunk software-pipelined
// lookahead, 20 coalesced b64 gathers in flight per chunk, branchless padding
// masks, non-temporal write-once output stores to preserve L2 table residency.
// Outer loop kept rolled (unroll 1) to stay inside the 106-SGPR file.

typedef __attribute__((ext_vector_type(2))) float v2f;
typedef __attribute__((ext_vector_type(4))) float v4f;

namespace {
constexpr int PAD      = 0;
constexpr int L_HIST   = 50;
constexpr int D_EMB    = 64;
constexpr int L_CTX    = 20;
constexpr int D_CTX    = 32;
constexpr int D_DENSE  = 128;
constexpr int OFF_CATE  = D_EMB;                    // 64
constexpr int OFF_CTX   = 2 * D_EMB;                // 128
constexpr int OFF_PRICE = OFF_CTX + L_CTX * D_CTX;  // 768
constexpr int OFF_DENSE = OFF_PRICE + 1;            // 769
constexpr int OUT_W     = OFF_DENSE + D_DENSE;      // 897
constexpr int WPB       = 8;                        // wave32s per block
constexpr int CHUNK     = 10;                       // gathers batched per table
constexpr int NCHUNK    = L_HIST / CHUNK;           // 5
}

__global__ __launch_bounds__(WPB * 32) void feature_layer_kernel(
    const int*   __restrict__ item_hist,   // [B, 50]
    const int*   __restrict__ cate_hist,   // [B, 50]
    const int*   __restrict__ ctx_ids,     // [B, 20]
    const float* __restrict__ price,       // [B, 1]
    const float* __restrict__ dense_vec,   // [B, 128]
    const float* __restrict__ W_item,      // [100000, 64]
    const float* __restrict__ W_cate,      // [100000, 64]
    const float* __restrict__ W_ctx,       // [1000, 32]
    float*       __restrict__ out,         // [B, 897]
    int B)
{
    const int lane = threadIdx.x & 31;
    int b = blockIdx.x * WPB + (threadIdx.x >> 5);
    if (b >= B) return;
    // b is wave-uniform; pin it to an SGPR so index reads become wide scalar
    // (SMEM) loads and VMEM slots are reserved for the embedding gathers.
    b = __builtin_amdgcn_readfirstlane(b);

    const int* __restrict__ ih = item_hist + b * L_HIST;
    const int* __restrict__ ch = cate_hist + b * L_HIST;
    const int* __restrict__ cx = ctx_ids   + b * L_CTX;
    float*     __restrict__ orow = out + (size_t)b * OUT_W;

    // Row-major v2f views of the tables: lane owns dims {2*lane, 2*lane+1},
    // so each gather is one fully-coalesced 256B global_load_b64.
    const v2f* __restrict__ Wi2 = reinterpret_cast<const v2f*>(W_item);
    const v2f* __restrict__ Wc2 = reinterpret_cast<const v2f*>(W_cate);

    float ai0 = 0.f, ai1 = 0.f, ac0 = 0.f, ac1 = 0.f;

    // Software pipeline: indices for chunk c+1 are fetched (SMEM) while the
    // gathers of chunk c are in flight, so the back-edge never stalls on
    // s_wait_kmcnt with an empty gather queue.
    int ii[CHUNK], ic[CHUNK];
    #pragma unroll
    for (int u = 0; u < CHUNK; ++u) {
        ii[u] = ih[u];
        ic[u] = ch[u];
    }

    #pragma unroll 1   // keep rolled: full unroll spills the SGPR file
    for (int c = 0; c < NCHUNK; ++c) {
        // 1) Issue all 2*CHUNK gathers before any use: deep LOADcnt batch.
        v2f vi[CHUNK], vc[CHUNK];
        #pragma unroll
        for (int u = 0; u < CHUNK; ++u) {
            vi[u] = Wi2[(size_t)ii[u] * (D_EMB / 2) + lane];
            vc[u] = Wc2[(size_t)ic[u] * (D_EMB / 2) + lane];
        }
        // 2) Snapshot padding masks before the index registers are recycled.
        float mi[CHUNK], mc[CHUNK];
        #pragma unroll
        for (int u = 0; u < CHUNK; ++u) {
            mi[u] = (ii[u] != PAD) ? 1.0f : 0.0f;
            mc[u] = (ic[u] != PAD) ? 1.0f : 0.0f;
        }
        // 3) Lookahead: fetch next chunk's indices (branchless clamp; the
        //    last iteration harmlessly re-reads the final chunk).
        const int nbase = (c + 1 < NCHUNK) ? (c + 1) * CHUNK : (NCHUNK - 1) * CHUNK;
        #pragma unroll
        for (int u = 0; u < CHUNK; ++u) {
            ii[u] = ih[nbase + u];
            ic[u] = ch[nbase + u];
        }
        // 4) Drain: masked FMA accumulation.
        #pragma unroll
        for (int u = 0; u < CHUNK; ++u) {
            ai0 = fmaf(mi[u], vi[u].x, ai0);
            ai1 = fmaf(mi[u], vi[u].y, ai1);
            ac0 = fmaf(mc[u], vc[u].x, ac0);
            ac1 = fmaf(mc[u], vc[u].y, ac1);
        }
    }
    // Write-once output (59 MB): non-temporal stores protect L2 residency of
    // the two 25.6 MB embedding tables.
    __builtin_nontemporal_store(ai0, orow + 2 * lane);
    __builtin_nontemporal_store(ai1, orow + 2 * lane + 1);
    __builtin_nontemporal_store(ac0 / (float)L_HIST, orow + OFF_CATE + 2 * lane);
    __builtin_nontemporal_store(ac1 / (float)L_HIST, orow + OFF_CATE + 2 * lane + 1);

    // ---- ctx: 20 indices up front, then 20 coalesced 128B gathers + NT stores.
    {
        int ix[L_CTX];
        #pragma unroll
        for (int l = 0; l < L_CTX; ++l) ix[l] = cx[l];

        float v[L_CTX];
        #pragma unroll
        for (int l = 0; l < L_CTX; ++l)
            v[l] = W_ctx[(size_t)ix[l] * D_CTX + lane];

        #pragma unroll
        for (int l = 0; l < L_CTX; ++l) {
            const float vv = (ix[l] != PAD) ? v[l] : 0.0f;
            __builtin_nontemporal_store(vv, orow + OFF_CTX + l * D_CTX + lane);
        }
    }

    // ---- price passthrough
    if (lane == 0) {
        __builtin_nontemporal_store(price[b], orow + OFF_PRICE);
    }

    // ---- dense passthrough: one b128 NT load per lane (source 16B aligned);
    // destination offset 769 is odd -> scalar stores (region is tiny, 8.4 MB).
    const v4f dv = __builtin_nontemporal_load(
        reinterpret_cast<const v4f*>(dense_vec + (size_t)b * D_DENSE) + lane);
    float* dd = orow + OFF_DENSE + 4 * lane;
    __builtin_nontemporal_store(dv.x, dd + 0);
    __builtin_nontemporal_store(dv.y, dd + 1);
    __builtin_nontemporal_store(dv.z, dd + 2);
    __builtin_nontemporal_store(dv.w, dd + 3);
}

extern "C" void kernel_launch(void* const* d_in, const int* in_sizes, int n_in,
                              void* d_out, int out_size, void* d_ws, size_t ws_size,
                              hipStream_t stream) {
    (void)n_in; (void)out_size; (void)d_ws; (void)ws_size;
    const int*   item_hist = (const int*)  d_in[0];
    const int*   cate_hist = (const int*)  d_in[1];
    const int*   ctx_ids   = (const int*)  d_in[2];
    const float* price     = (const float*)d_in[3];
    const float* dense_vec = (const float*)d_in[4];
    const float* W_item    = (const float*)d_in[5];
    const float* W_cate    = (const float*)d_in[6];
    const float* W_ctx     = (const float*)d_in[7];
    float* out = (float*)d_out;

    const int B = in_sizes[3];  // price: one element per row
    dim3 grid((B + WPB - 1) / WPB);
    dim3 block(WPB * 32);
    feature_layer_kernel<<<grid, block, 0, stream>>>(
        item_hist, cate_hist, ctx_ids, price, dense_vec,
        W_item, W_cate, W_ctx, out, B);
}